// GCNFG_35081292874128
// MI455X (gfx1250) — compile-verified
//
#include <hip/hip_runtime.h>
#include <hip/hip_bf16.h>
#include <math.h>

// ---------------- model dimensions ----------------
#define NNODES 100000
#define NEDGES 400000
#define BSZ    1024
#define DXD    78
#define SEQ    1000
#define EMB    128
#define HID    300
#define NF     32
#define OUTD   128
#define CONVT  121          // 128 - 8 + 1

typedef float v8f __attribute__((ext_vector_type(8)));
typedef float v2f __attribute__((ext_vector_type(2)));

#define WMMA_F32(a, b, c) \
    __builtin_amdgcn_wmma_f32_16x16x4_f32(false, (a), false, (b), (short)0, (c), false, false)

// =======================================================================
// fp32 GEMM on the CDNA5 matrix pipe: C[M,N] = A[M,K] @ B[K,N] (+bias, relu)
// One wave computes a 32x32 C tile = 2x2 grid of 16x16 WMMA tiles (4 accums).
// CONTRACT: M % 32 == 0 (true for all call sites: 100000, 13312, 1024).
// N is handled by clamped loads (no predication in the hot loop) + guarded
// stores. K tail handled by value-zeroing the A fragment (v_cndmask, not
// exec branches).
// Fragment layouts per ISA 7.12.2:
//   A (16x4): lanes 0-15 row=lane, K={k0,k0+1}; lanes 16-31 K={k0+2,k0+3}
//   B (4x16): lanes 0-15 col=lane, same K split
//   C/D (16x16): VGPR r -> row r (lanes 0-15) / row 8+r (lanes 16-31), col=lane&15
// =======================================================================
__global__ __launch_bounds__(32) void gemm_f32_wmma(
    const float* __restrict__ A, int lda,
    const float* __restrict__ Bw, int ldb,
    const float* __restrict__ bias,
    float* __restrict__ C, int ldc,
    int M, int N, int K, int relu)
{
    const int lane = threadIdx.x;
    const int half = lane >> 4;          // 0: K pair {0,1}; 1: K pair {2,3}
    const int l16  = lane & 15;
    const int koff = half * 2;

    const int am0 = blockIdx.y * 32 + l16;       // A row, tile row 0
    const int am1 = am0 + 16;                    // A row, tile row 1 (in-bounds: M%32==0)
    const int bnA = blockIdx.x * 32 + l16;       // B col, tile col 0
    const int bnB = bnA + 16;                    // B col, tile col 1
    const int bnAl = min(bnA, N - 1);            // clamped (always loadable)
    const int bnBl = min(bnB, N - 1);

    const float* __restrict__ Ap0 = A + (size_t)am0 * lda + koff;
    const float* __restrict__ Ap1 = A + (size_t)am1 * lda + koff;
    const float* __restrict__ BpA = Bw + (size_t)koff * ldb + bnAl;
    const float* __restrict__ BpB = Bw + (size_t)koff * ldb + bnBl;

    v8f acc00 = {}, acc01 = {}, acc10 = {}, acc11 = {};
    const int Kmain = K & ~3;

    #pragma unroll 2
    for (int k0 = 0; k0 < Kmain; k0 += 4) {
        v2f a0, a1, b0, b1;
        a0.x = Ap0[k0];     a0.y = Ap0[k0 + 1];
        a1.x = Ap1[k0];     a1.y = Ap1[k0 + 1];
        const size_t r0 = (size_t)k0 * ldb;
        const size_t r1 = r0 + ldb;
        b0.x = BpA[r0];     b0.y = BpA[r1];
        b1.x = BpB[r0];     b1.y = BpB[r1];
        acc00 = WMMA_F32(a0, b0, acc00);
        acc01 = WMMA_F32(a0, b1, acc01);
        acc10 = WMMA_F32(a1, b0, acc10);
        acc11 = WMMA_F32(a1, b1, acc11);
    }
    if (Kmain < K) {                       // K % 4 tail (K=78 -> 2, K=133 -> 1)
        const int ka  = Kmain + koff;
        const float m0 = (ka     < K) ? 1.0f : 0.0f;   // value masks -> v_cndmask
        const float m1 = (ka + 1 < K) ? 1.0f : 0.0f;
        const int ka0 = min(ka,     K - 1);            // clamped, always loadable
        const int ka1 = min(ka + 1, K - 1);
        v2f a0, a1, b0, b1;
        a0.x = A[(size_t)am0 * lda + ka0] * m0;  a0.y = A[(size_t)am0 * lda + ka1] * m1;
        a1.x = A[(size_t)am1 * lda + ka0] * m0;  a1.y = A[(size_t)am1 * lda + ka1] * m1;
        b0.x = Bw[(size_t)ka0 * ldb + bnAl];     b0.y = Bw[(size_t)ka1 * ldb + bnAl];
        b1.x = Bw[(size_t)ka0 * ldb + bnBl];     b1.y = Bw[(size_t)ka1 * ldb + bnBl];
        acc00 = WMMA_F32(a0, b0, acc00);
        acc01 = WMMA_F32(a0, b1, acc01);
        acc10 = WMMA_F32(a1, b0, acc10);
        acc11 = WMMA_F32(a1, b1, acc11);
    }

    // ---- store 32x32 tile ----
    const int rb = blockIdx.y * 32 + half * 8;
    float bvA = 0.0f, bvB = 0.0f;
    if (bias) { bvA = bias[bnAl]; bvB = bias[bnBl]; }   // clamped, unconditional
    if (bnA < N) {
        #pragma unroll
        for (int r = 0; r < 8; ++r) {
            float v0 = acc00[r] + bvA; if (relu) v0 = fmaxf(v0, 0.0f);
            C[(size_t)(rb + r) * ldc + bnA] = v0;
            float v1 = acc10[r] + bvA; if (relu) v1 = fmaxf(v1, 0.0f);
            C[(size_t)(rb + 16 + r) * ldc + bnA] = v1;
        }
    }
    if (bnB < N) {
        #pragma unroll
        for (int r = 0; r < 8; ++r) {
            float v0 = acc01[r] + bvB; if (relu) v0 = fmaxf(v0, 0.0f);
            C[(size_t)(rb + r) * ldc + bnB] = v0;
            float v1 = acc11[r] + bvB; if (relu) v1 = fmaxf(v1, 0.0f);
            C[(size_t)(rb + 16 + r) * ldc + bnB] = v1;
        }
    }
}

static inline void launch_gemm(const float* A, int lda, const float* B, int ldb,
                               const float* bias, float* C, int ldc,
                               int M, int N, int K, int relu, hipStream_t s)
{
    dim3 grid((N + 31) / 32, M / 32);     // M % 32 == 0 contract
    gemm_f32_wmma<<<grid, 32, 0, s>>>(A, lda, B, ldb, bias, C, ldc, M, N, K, relu);
}

// ---------------- elementwise / graph kernels ----------------
__global__ void zero_kernel(float* __restrict__ p, size_t n)
{
    size_t i = (size_t)blockIdx.x * blockDim.x + threadIdx.x;
    size_t stride = (size_t)gridDim.x * blockDim.x;
    for (; i < n; i += stride) p[i] = 0.0f;
}

__global__ void deg_kernel(const int* __restrict__ dst, float* __restrict__ deg)
{
    int i = blockIdx.x * blockDim.x + threadIdx.x;
    if (i < NEDGES) atomicAdd(&deg[dst[i]], 1.0f);
}

__global__ void dis_kernel(float* __restrict__ d)
{
    int i = blockIdx.x * blockDim.x + threadIdx.x;
    if (i < NNODES) d[i] = rsqrtf(d[i] + 1.0f);
}

// agg[n,d] = hw[n,d]*dis[n]^2 + bias[d]
__global__ void gcn_init_kernel(const float* __restrict__ hw, const float* __restrict__ dis,
                                const float* __restrict__ bias, float* __restrict__ agg, int D)
{
    size_t idx = (size_t)blockIdx.x * blockDim.x + threadIdx.x;
    size_t tot = (size_t)NNODES * D;
    if (idx >= tot) return;
    int n = (int)(idx / D), d = (int)(idx % D);
    float s = dis[n];
    agg[idx] = hw[idx] * s * s + bias[d];
}

// agg[dst,d] += hw[src,d]*dis[src]*dis[dst]
__global__ void gcn_edge_kernel(const float* __restrict__ hw, const float* __restrict__ dis,
                                const int* __restrict__ src, const int* __restrict__ dst,
                                float* __restrict__ agg, int D)
{
    size_t idx = (size_t)blockIdx.x * blockDim.x + threadIdx.x;
    size_t tot = (size_t)NEDGES * D;
    if (idx >= tot) return;
    int e = (int)(idx / D), d = (int)(idx % D);
    int s = src[e], t = dst[e];
    atomicAdd(&agg[(size_t)t * D + d], hw[(size_t)s * D + d] * dis[s] * dis[t]);
}

__global__ void relu_kernel(float* __restrict__ p, size_t n)
{
    size_t i = (size_t)blockIdx.x * blockDim.x + threadIdx.x;
    if (i < n) p[i] = fmaxf(p[i], 0.0f);
}

// global max pool; h >= 0 post-ReLU so uint bit-compare == float compare; init 0 covers empty segments
__global__ void pool_kernel(const float* __restrict__ h, const int* __restrict__ batch,
                            float* __restrict__ g, int D)
{
    size_t idx = (size_t)blockIdx.x * blockDim.x + threadIdx.x;
    size_t tot = (size_t)NNODES * D;
    if (idx >= tot) return;
    int n = (int)(idx / D), d = (int)(idx % D);
    unsigned bits = __float_as_uint(h[idx]);
    atomicMax((unsigned int*)&g[(size_t)batch[n] * D + d], bits);
}

// ---------------- protein conv branch (factorized) ----------------
// M[b,f,a,k] = sum_c [target[b,c]==a] * conv_w[f,c,k]   (layout [B][F*26*8])
__global__ __launch_bounds__(256) void build_M_kernel(const int* __restrict__ target,
                                                      const float* __restrict__ conv_w,
                                                      float* __restrict__ Mout)
{
    const int b = blockIdx.x;
    const int t = threadIdx.x;          // 256 = 32 filters x 8 taps
    const int f = t >> 3, kk = t & 7;
    __shared__ float acc[NF * 26 * 8];  // 6656 floats
    for (int i = t; i < NF * 26 * 8; i += 256) acc[i] = 0.0f;
    __syncthreads();
    const float* wfk = conv_w + ((size_t)f * SEQ) * 8 + kk;
    const int*   tg  = target + (size_t)b * SEQ;
    float* row = acc + (f * 26) * 8 + kk;     // exclusive (f,kk) slice
    for (int c = 0; c < SEQ; ++c) {
        int a = tg[c];
        row[a * 8] += wfk[(size_t)c * 8];
    }
    __syncthreads();
    float* Mo = Mout + (size_t)b * (NF * 26 * 8);
    for (int i = t; i < NF * 26 * 8; i += 256) Mo[i] = acc[i];
}

// out[b, f*121 + t] = conv_b[f] + sum_{a,k} M[b,f,a,k] * emb_xt[a, t+k]
__global__ void conv_out_kernel(const float* __restrict__ Mbuf, const float* __restrict__ emb_xt,
                                const float* __restrict__ conv_b, float* __restrict__ out)
{
    int idx = blockIdx.x * blockDim.x + threadIdx.x;
    const int tot = BSZ * NF * CONVT;
    if (idx >= tot) return;
    int tpos = idx % CONVT; int tmp = idx / CONVT;
    int f = tmp % NF; int b = tmp / NF;
    const float* Mb = Mbuf + (size_t)b * (NF * 26 * 8) + f * 26 * 8;
    float acc = conv_b[f];
    for (int a = 0; a < 26; ++a) {
        const float* e = emb_xt + a * EMB + tpos;
        const float* m = Mb + a * 8;
        #pragma unroll
        for (int kk = 0; kk < 8; ++kk) acc += m[kk] * e[kk];
    }
    out[(size_t)b * (NF * CONVT) + f * CONVT + tpos] = acc;
}

// ---------------- FG attention + LayerNorm (row 0 only) ----------------
__global__ __launch_bounds__(128) void attn_ln_kernel(
    const float* __restrict__ f, const float* __restrict__ q,
    const float* __restrict__ k, const float* __restrict__ v,
    const float* __restrict__ ln_g, const float* __restrict__ ln_b,
    float* __restrict__ out)
{
    const int b = blockIdx.x;
    const int t = threadIdx.x;        // 128
    __shared__ float red[13][128];
    __shared__ float p[13];
    __shared__ float fv[HID];
    __shared__ float stats[2];
    const size_t base = (size_t)b * 13 * HID;

    float part[13];
    #pragma unroll
    for (int j = 0; j < 13; ++j) part[j] = 0.0f;
    for (int d = t; d < HID; d += 128) {
        float qd = q[base + d];
        #pragma unroll
        for (int j = 0; j < 13; ++j) part[j] += qd * k[base + (size_t)j * HID + d];
    }
    #pragma unroll
    for (int j = 0; j < 13; ++j) red[j][t] = part[j];
    __syncthreads();
    if (t == 0) {
        float s[13]; float mx = -1e30f;
        const float scale = rsqrtf((float)HID);
        for (int j = 0; j < 13; ++j) {
            float a = 0.0f;
            for (int i = 0; i < 128; ++i) a += red[j][i];
            a *= scale; s[j] = a; if (a > mx) mx = a;
        }
        float se = 0.0f;
        for (int j = 0; j < 13; ++j) { s[j] = __expf(s[j] - mx); se += s[j]; }
        float inv = 1.0f / se;
        for (int j = 0; j < 13; ++j) p[j] = s[j] * inv;
    }
    __syncthreads();

    float lsum = 0.0f, lsq = 0.0f;
    for (int d = t; d < HID; d += 128) {
        float a = f[base + d];
        #pragma unroll
        for (int j = 0; j < 13; ++j) a += p[j] * v[base + (size_t)j * HID + d];
        fv[d] = a; lsum += a; lsq += a * a;
    }
    red[0][t] = lsum; red[1][t] = lsq;
    __syncthreads();
    if (t == 0) {
        float a = 0.0f, bq = 0.0f;
        for (int i = 0; i < 128; ++i) { a += red[0][i]; bq += red[1][i]; }
        float mu = a / (float)HID;
        float var = bq / (float)HID - mu * mu;
        stats[0] = mu; stats[1] = rsqrtf(var + 1e-5f);
    }
    __syncthreads();
    const float mu = stats[0], inv = stats[1];
    for (int d = t; d < HID; d += 128)
        out[(size_t)b * HID + d] = (fv[d] - mu) * inv * ln_g[d] + ln_b[d];
}

// ---------------- final 512 -> 1 dot ----------------
__global__ __launch_bounds__(128) void final_kernel(const float* __restrict__ t2,
                                                    const float* __restrict__ w,
                                                    const float* __restrict__ bias,
                                                    float* __restrict__ out)
{
    const int b = blockIdx.x;
    const int t = threadIdx.x;
    __shared__ float red[128];
    float a = 0.0f;
    for (int d = t; d < 512; d += 128) a += t2[(size_t)b * 512 + d] * w[d];
    red[t] = a;
    __syncthreads();
    for (int s = 64; s > 0; s >>= 1) { if (t < s) red[t] += red[t + s]; __syncthreads(); }
    if (t == 0) out[b] = red[0] + bias[0];
}

// =======================================================================
extern "C" void kernel_launch(void* const* d_in, const int* in_sizes, int n_in,
                              void* d_out, int out_size, void* d_ws, size_t ws_size,
                              hipStream_t stream)
{
    const float* x        = (const float*)d_in[0];
    const int*   eidx     = (const int*)  d_in[1];
    const int*   batch    = (const int*)  d_in[2];
    const int*   target   = (const int*)  d_in[3];
    const float* embs     = (const float*)d_in[4];
    const float* W1 = (const float*)d_in[5];  const float* b1 = (const float*)d_in[6];
    const float* W2 = (const float*)d_in[7];  const float* b2 = (const float*)d_in[8];
    const float* W3 = (const float*)d_in[9];  const float* b3 = (const float*)d_in[10];
    const float* fc_g1_w = (const float*)d_in[11]; const float* fc_g1_b = (const float*)d_in[12];
    const float* fc_g2_w = (const float*)d_in[13]; const float* fc_g2_b = (const float*)d_in[14];
    const float* emb_xt  = (const float*)d_in[15];
    const float* conv_w  = (const float*)d_in[16]; const float* conv_b  = (const float*)d_in[17];
    const float* fc1xt_w = (const float*)d_in[18]; const float* fc1xt_b = (const float*)d_in[19];
    const float* fg_lin_w= (const float*)d_in[20]; const float* fg_lin_b= (const float*)d_in[21];
    const float* Wq = (const float*)d_in[22];
    const float* Wk = (const float*)d_in[23];
    const float* Wv = (const float*)d_in[24];
    const float* ln_g = (const float*)d_in[25]; const float* ln_b = (const float*)d_in[26];
    const float* fg_out_w= (const float*)d_in[27]; const float* fg_out_b= (const float*)d_in[28];
    const float* fc1_w = (const float*)d_in[29]; const float* fc1_b = (const float*)d_in[30];
    const float* fc2_w = (const float*)d_in[31]; const float* fc2_b = (const float*)d_in[32];
    const float* out_w = (const float*)d_in[33]; const float* out_b = (const float*)d_in[34];

    const int* src = eidx;
    const int* dst = eidx + NEDGES;

    // ---- workspace layout (floats) ----
    float* ws = (float*)d_ws;
    const size_t P = (size_t)NNODES * 312;      // 31.2M floats per node buffer
    float* P0   = ws;
    float* P1   = ws + P;
    float* P2   = ws + 2 * P;
    float* dis  = ws + 3 * P;                   // 100000
    float* g    = dis + 100032;                 // 1024 x 312
    float* t1024= g + (size_t)BSZ * 312;        // 1024 x 1024
    float* xc   = t1024 + (size_t)BSZ * 1024;   // 1024 x 384
    float* Mbuf = xc + (size_t)BSZ * 384;       // 1024 x 6656
    float* xtconv = Mbuf + (size_t)BSZ * (NF * 26 * 8);   // 1024 x 3872
    // aliases: P0 free after GCN, P2 free after layer-3 GEMM
    float* fbuf  = P0;
    float* qbuf  = P0 + (size_t)4  * 1024 * 1024;
    float* kbuf  = P0 + (size_t)8  * 1024 * 1024;
    float* vbuf  = P0 + (size_t)12 * 1024 * 1024;
    float* fgrow = P0 + (size_t)16 * 1024 * 1024;   // 1024 x 300
    float* t1 = P2;                                 // 1024 x 1024
    float* t2 = P2 + (size_t)BSZ * 1024;            // 1024 x 512

    // ---- degree / normalization ----
    zero_kernel<<<512, 256, 0, stream>>>(dis, (size_t)NNODES);
    deg_kernel<<<(NEDGES + 255) / 256, 256, 0, stream>>>(dst, dis);
    dis_kernel<<<(NNODES + 255) / 256, 256, 0, stream>>>(dis);

    // ---- GCN layer 1: 78 -> 78 ----
    launch_gemm(x, DXD, W1, DXD, nullptr, P0, DXD, NNODES, DXD, DXD, 0, stream);
    {
        size_t tot = (size_t)NNODES * DXD;
        gcn_init_kernel<<<(unsigned)((tot + 255) / 256), 256, 0, stream>>>(P0, dis, b1, P1, DXD);
        size_t te = (size_t)NEDGES * DXD;
        gcn_edge_kernel<<<(unsigned)((te + 255) / 256), 256, 0, stream>>>(P0, dis, src, dst, P1, DXD);
        relu_kernel<<<(unsigned)((tot + 255) / 256), 256, 0, stream>>>(P1, tot);
    }
    // ---- GCN layer 2: 78 -> 156 ----
    launch_gemm(P1, DXD, W2, 2 * DXD, nullptr, P0, 2 * DXD, NNODES, 2 * DXD, DXD, 0, stream);
    {
        const int D = 2 * DXD;
        size_t tot = (size_t)NNODES * D;
        gcn_init_kernel<<<(unsigned)((tot + 255) / 256), 256, 0, stream>>>(P0, dis, b2, P2, D);
        size_t te = (size_t)NEDGES * D;
        gcn_edge_kernel<<<(unsigned)((te + 255) / 256), 256, 0, stream>>>(P0, dis, src, dst, P2, D);
        relu_kernel<<<(unsigned)((tot + 255) / 256), 256, 0, stream>>>(P2, tot);
    }
    // ---- GCN layer 3: 156 -> 312 ----
    launch_gemm(P2, 2 * DXD, W3, 4 * DXD, nullptr, P0, 4 * DXD, NNODES, 4 * DXD, 2 * DXD, 0, stream);
    {
        const int D = 4 * DXD;
        size_t tot = (size_t)NNODES * D;
        gcn_init_kernel<<<(unsigned)((tot + 255) / 256), 256, 0, stream>>>(P0, dis, b3, P1, D);
        size_t te = (size_t)NEDGES * D;
        gcn_edge_kernel<<<(unsigned)((te + 255) / 256), 256, 0, stream>>>(P0, dis, src, dst, P1, D);
        relu_kernel<<<(unsigned)((tot + 255) / 256), 256, 0, stream>>>(P1, tot);
    }

    // ---- global max pool + graph head ----
    zero_kernel<<<512, 256, 0, stream>>>(g, (size_t)BSZ * 312);
    {
        size_t tot = (size_t)NNODES * 312;
        pool_kernel<<<(unsigned)((tot + 255) / 256), 256, 0, stream>>>(P1, batch, g, 312);
    }
    launch_gemm(g, 312, fc_g1_w, 1024, fc_g1_b, t1024, 1024, BSZ, 1024, 312, 1, stream);
    launch_gemm(t1024, 1024, fc_g2_w, OUTD, fc_g2_b, xc + 0, 3 * OUTD, BSZ, OUTD, 1024, 0, stream);

    // ---- protein conv branch (factorized conv) ----
    build_M_kernel<<<BSZ, 256, 0, stream>>>(target, conv_w, Mbuf);
    conv_out_kernel<<<(BSZ * NF * CONVT + 255) / 256, 256, 0, stream>>>(Mbuf, emb_xt, conv_b, xtconv);
    launch_gemm(xtconv, NF * CONVT, fc1xt_w, OUTD, fc1xt_b, xc + OUTD, 3 * OUTD,
                BSZ, OUTD, NF * CONVT, 0, stream);

    // ---- functional-group branch ----
    launch_gemm(embs, 133, fg_lin_w, HID, fg_lin_b, fbuf, HID, BSZ * 13, HID, 133, 0, stream);
    launch_gemm(fbuf, HID, Wq, HID, nullptr, qbuf, HID, BSZ * 13, HID, HID, 0, stream);
    launch_gemm(fbuf, HID, Wk, HID, nullptr, kbuf, HID, BSZ * 13, HID, HID, 0, stream);
    launch_gemm(fbuf, HID, Wv, HID, nullptr, vbuf, HID, BSZ * 13, HID, HID, 0, stream);
    attn_ln_kernel<<<BSZ, 128, 0, stream>>>(fbuf, qbuf, kbuf, vbuf, ln_g, ln_b, fgrow);
    launch_gemm(fgrow, HID, fg_out_w, OUTD, fg_out_b, xc + 2 * OUTD, 3 * OUTD,
                BSZ, OUTD, HID, 0, stream);

    // ---- head MLP ----
    launch_gemm(xc, 3 * OUTD, fc1_w, 1024, fc1_b, t1, 1024, BSZ, 1024, 3 * OUTD, 1, stream);
    launch_gemm(t1, 1024, fc2_w, 512, fc2_b, t2, 512, BSZ, 512, 1024, 1, stream);
    final_kernel<<<BSZ, 128, 0, stream>>>(t2, out_w, out_b, (float*)d_out);
}